// YoloLoss_37915971289865
// MI455X (gfx1250) — compile-verified
//
#include <hip/hip_runtime.h>

typedef __attribute__((ext_vector_type(2))) float v2f;
typedef __attribute__((ext_vector_type(8))) float v8f;

#define YL_COORD 5.0f
#define YL_NOOBJ 0.5f
#define YL_INV_S (1.0f / 7.0f)

// Reduce 32 per-lane partials to a single total using one
// v_wmma_f32_16x16x4_f32:  A = ones(16x4), B holds each lane's partial in
// exactly one of the 64 B-matrix slots (VGPR1 of B is zero).  Then every
// element of D = A*B is a column-sum of B, and summing D row 0's 16 columns
// (4 xor-shuffle adds) yields the grand total in every lane.
// Requires EXEC == all ones (call only from a fully-active wave).
__device__ __forceinline__ float wave_reduce_wmma(float p) {
    v2f a; a[0] = 1.0f; a[1] = 1.0f;   // A-matrix: all ones
    v2f b; b[0] = p;    b[1] = 0.0f;   // B-matrix: one partial per lane
    v8f c = {};
    c = __builtin_amdgcn_wmma_f32_16x16x4_f32(
        /*neg_a=*/false, a, /*neg_b=*/false, b,
        /*c_mod=*/(short)0, c, /*reuse_a=*/false, /*reuse_b=*/false);
    float v = c[0];
    v += __shfl_xor(v, 1, 32);
    v += __shfl_xor(v, 2, 32);
    v += __shfl_xor(v, 4, 32);
    v += __shfl_xor(v, 8, 32);
    return v;   // every lane now holds the total
}

__device__ __forceinline__ float iou_f(float b0, float b1, float b2, float b3,
                                       float l0, float l1, float l2, float l3) {
    float x1 = b0 * YL_INV_S - b2 * 0.5f;
    float y1 = b1 * YL_INV_S - b3 * 0.5f;
    float x2 = b0 * YL_INV_S + b2 * 0.5f;
    float y2 = b1 * YL_INV_S + b3 * 0.5f;
    float u1 = l0 * YL_INV_S - l2 * 0.5f;
    float v1 = l1 * YL_INV_S - l3 * 0.5f;
    float u2 = l0 * YL_INV_S + l2 * 0.5f;
    float v2 = l1 * YL_INV_S + l3 * 0.5f;
    float a1 = (x2 - x1) * (y2 - y1);
    float a2 = (u2 - u1) * (v2 - v1);
    float left   = fmaxf(x1, u1);
    float right  = fminf(x2, u2);
    float top    = fmaxf(y1, v1);
    float bottom = fminf(y2, v2);
    bool valid = (left < right) && (top < bottom);
    float inter = valid ? (right - left) * (bottom - top) : 0.0f;
    float uni = a1 + a2 - inter;
    return valid ? (inter / uni) : 0.0f;
}

// One thread per grid cell: stream 2x104B as coalesced b64 loads, compute the
// cell's loss contribution in registers, block-reduce via LDS + one WMMA.
__global__ void __launch_bounds__(256)
yolo_cell_kernel(const float* __restrict__ pred, const float* __restrict__ lab,
                 float* __restrict__ partial, int nCells) {
    const int tid = threadIdx.x;
    const int n = blockIdx.x * 256 + tid;

    float contrib = 0.0f;
    if (n < nCells) {
        float P[26], L[26];
        const float2* p2 = reinterpret_cast<const float2*>(pred) + (size_t)n * 13;
        const float2* l2 = reinterpret_cast<const float2*>(lab)  + (size_t)n * 13;
#pragma unroll
        for (int i = 0; i < 13; ++i) { float2 t = p2[i]; P[2*i] = t.x; P[2*i+1] = t.y; }
#pragma unroll
        for (int i = 0; i < 13; ++i) { float2 t = l2[i]; L[2*i] = t.x; L[2*i+1] = t.y; }

        const float conf    = L[4];
        const float coord_m = (conf > 0.0f)  ? 1.0f : 0.0f;
        const float noobj_m = (conf == 0.0f) ? 1.0f : 0.0f;
        const float c1 = P[4], c2 = P[9];

        float noobj_c = noobj_m * (c1 * c1 + c2 * c2);

        float cls = 0.0f;
#pragma unroll
        for (int k = 10; k < 26; ++k) { float d = P[k] - L[k]; cls += d * d; }

        const float iou1 = iou_f(P[0], P[1], P[2], P[3], L[0], L[1], L[2], L[3]);
        const float iou2 = iou_f(P[5], P[6], P[7], P[8], L[0], L[1], L[2], L[3]);
        const bool pick1 = iou1 > iou2;

        const float s0 = pick1 ? P[0] : P[5];
        const float s1 = pick1 ? P[1] : P[6];
        const float s2 = pick1 ? P[2] : P[7];
        const float s3 = pick1 ? P[3] : P[8];

        const float d0 = s0 - L[0];
        const float d1 = s1 - L[1];
        const float q2 = sqrtf(s2) - sqrtf(L[2]);
        const float q3 = sqrtf(s3) - sqrtf(L[3]);
        const float coord_cell = d0 * d0 + d1 * d1 + q2 * q2 + q3 * q3;

        const float dc = pick1 ? (c1 - iou1) : (c2 - iou2);
        const float obj_c = dc * dc;
        const float ne = pick1 ? c2 : c1;
        const float noobj_extra = ne * ne;

        contrib = YL_COORD * coord_m * coord_cell
                + coord_m * obj_c
                + YL_NOOBJ * (noobj_c + coord_m * noobj_extra)
                + coord_m * cls;
    }

    __shared__ float smem[256];
    smem[tid] = contrib;
    __syncthreads();

    if (tid < 32) {                       // exactly wave 0, all 32 lanes active
        float q = smem[tid];
#pragma unroll
        for (int i = 1; i < 8; ++i) q += smem[tid + 32 * i];
        float tot = wave_reduce_wmma(q);  // EXEC all-ones here
        if (tid == 0) partial[blockIdx.x] = tot;
    }
}

// Single-wave deterministic final reduction (no float atomics anywhere).
__global__ void __launch_bounds__(32)
yolo_final_kernel(const float* __restrict__ partial, int nBlocks, float invN,
                  float* __restrict__ out) {
    const int lane = threadIdx.x;  // one wave32, EXEC all ones
    float q = 0.0f;
    for (int i = lane; i < nBlocks; i += 32) q += partial[i];
    float tot = wave_reduce_wmma(q);
    if (lane == 0) out[0] = tot * invN;
}

extern "C" void kernel_launch(void* const* d_in, const int* in_sizes, int n_in,
                              void* d_out, int out_size, void* d_ws, size_t ws_size,
                              hipStream_t stream) {
    const float* pred = (const float*)d_in[0];
    const float* lab  = (const float*)d_in[1];
    float* out        = (float*)d_out;
    float* partial    = (float*)d_ws;

    const int total  = in_sizes[0];          // B*7*7*26
    const int nCells = total / 26;           // B*49
    const int nBatch = nCells / 49;          // B
    const int blocks = (nCells + 255) / 256;

    yolo_cell_kernel<<<blocks, 256, 0, stream>>>(pred, lab, partial, nCells);
    yolo_final_kernel<<<1, 32, 0, stream>>>(partial, blocks, 1.0f / (float)nBatch, out);
}